// FundamentalMatrixEstimator_2783138808070
// MI455X (gfx1250) — compile-verified
//
#include <hip/hip_runtime.h>
#include <hip/hip_bf16.h>

typedef __attribute__((ext_vector_type(8))) float v8f;
typedef __attribute__((ext_vector_type(2))) float v2f;

#define NPTS 2048
#define DIM  9
#define NUM_ITERS 5
#define POW_ITERS 24

#define WMMA_F32X4(a, b, c)                                            \
    __builtin_amdgcn_wmma_f32_16x16x4_f32(false, (a), false, (b),      \
                                          (short)0, (c), false, false)

// Fused pass over one batch's A [2048 x 9]:
//  - (WEIGHTED) residual r = a_n . f via intra-half shfl reduction
//  - Geman-McClure weight w = 2/(1+r^2)^2
//  - accumulate M += w * a_n a_n^T with V_WMMA_F32_16X16X4_F32
// 8 rows / iteration into two independent accumulators. Padding lanes
// (m = 9..15) load a clamped in-bounds column; their values only touch
// rows/cols 9..15 of the padded 16x16 result (never read), and their
// residual contribution is zero because fm == 0 there -> no exec branching.
template <bool WEIGHTED>
__device__ inline v8f accumulate_gram(const float* __restrict__ Ab, int lane, float fm)
{
    const int half = lane >> 4;
    const int m    = lane & 15;
    const int mc   = (m < DIM) ? m : (DIM - 1);   // clamped column

    v8f c0 = {}, c1 = {};
    // lane covers rows (8t + 2*half) / (+1) and (8t + 4 + 2*half) / (+1), col mc
    const float* p = Ab + half * (2 * DIM) + mc;

    #pragma unroll 1
    for (int t = 0; t < NPTS / 8; ++t) {
        const float bx0 = p[0];            // row 8t+2h     , col mc
        const float by0 = p[DIM];          // row 8t+2h + 1 , col mc
        const float bx1 = p[4 * DIM];      // row 8t+4+2h   , col mc
        const float by1 = p[5 * DIM];      // row 8t+4+2h+1 , col mc
        __builtin_prefetch(p + 8 * DIM * 8, 0, 1);   // ~2.3KB ahead

        float wx0 = 1.0f, wy0 = 1.0f, wx1 = 1.0f, wy1 = 1.0f;
        if (WEIGHTED) {
            float px0 = bx0 * fm, py0 = by0 * fm;    // fm==0 on padding lanes
            float px1 = bx1 * fm, py1 = by1 * fm;
            // sum over the 16 lanes of each half (masks < 16 stay in-half)
            px0 += __shfl_xor(px0, 1);  py0 += __shfl_xor(py0, 1);
            px1 += __shfl_xor(px1, 1);  py1 += __shfl_xor(py1, 1);
            px0 += __shfl_xor(px0, 2);  py0 += __shfl_xor(py0, 2);
            px1 += __shfl_xor(px1, 2);  py1 += __shfl_xor(py1, 2);
            px0 += __shfl_xor(px0, 4);  py0 += __shfl_xor(py0, 4);
            px1 += __shfl_xor(px1, 4);  py1 += __shfl_xor(py1, 4);
            px0 += __shfl_xor(px0, 8);  py0 += __shfl_xor(py0, 8);
            px1 += __shfl_xor(px1, 8);  py1 += __shfl_xor(py1, 8);
            // w = MAX_W / (1 + (r/K)^2)^2 ; K=1, MAX_W=2 ; r only squared
            const float t0 = 1.0f + px0 * px0, u0 = 1.0f + py0 * py0;
            const float t1 = 1.0f + px1 * px1, u1 = 1.0f + py1 * py1;
            wx0 = 2.0f * __builtin_amdgcn_rcpf(t0 * t0);
            wy0 = 2.0f * __builtin_amdgcn_rcpf(u0 * u0);
            wx1 = 2.0f * __builtin_amdgcn_rcpf(t1 * t1);
            wy1 = 2.0f * __builtin_amdgcn_rcpf(u1 * u1);
        }

        v2f a0, b0, a1, b1;
        b0.x = bx0;       b0.y = by0;         // B(k,j) = A[row k][j]
        a0.x = bx0 * wx0; a0.y = by0 * wy0;   // A(i,k) = w_k * A[row k][i]
        b1.x = bx1;       b1.y = by1;
        a1.x = bx1 * wx1; a1.y = by1 * wy1;
        c0 = WMMA_F32X4(a0, b0, c0);          // two independent accum chains
        c1 = WMMA_F32X4(a1, b1, c1);
        p += 8 * DIM;
    }
    return c0 + c1;
}

// Dump the 9x9 live part of the 16x16 WMMA accumulator into LDS (stride 16).
// Accumulator layout: VGPR r holds M(i=r+8*half, j=m) for lane(half,m).
__device__ inline void dump_accum(const v8f& c, float* lds_M, int lane)
{
    const int half = lane >> 4;
    const int m    = lane & 15;
    if (m < DIM) {
        #pragma unroll
        for (int r = 0; r < 8; ++r) {
            const int i = r + 8 * half;
            if (half == 0 || r == 0)     // i = 0..7 (half 0), i = 8 (half 1)
                lds_M[i * 16 + m] = c[r];
        }
    }
}

// All lanes redundantly: Cholesky of (M + eps*I), then inverse power
// iteration -> eigenvector of the smallest eigenvalue. Fully unrolled so all
// array indexing is static (stays in registers / gets SALU-scalarized).
__device__ inline void solve_min_eigvec(const float* lds_M, float* lds_f, int lane)
{
    float L[DIM][DIM];
    #pragma unroll
    for (int i = 0; i < DIM; ++i) {
        #pragma unroll
        for (int j = 0; j <= i; ++j)
            L[i][j] = lds_M[i * 16 + j];
        L[i][i] += 1e-10f;           // EIGH_EPS augmentation
    }

    float dinv[DIM];
    #pragma unroll
    for (int j = 0; j < DIM; ++j) {
        float d = L[j][j];
        #pragma unroll
        for (int k = 0; k < j; ++k) d -= L[j][k] * L[j][k];
        d = d > 1e-30f ? d : 1e-30f;
        const float s = __builtin_sqrtf(d);
        L[j][j] = s;
        dinv[j] = __builtin_amdgcn_rcpf(s);
        #pragma unroll
        for (int i = j + 1; i < DIM; ++i) {
            float v = L[i][j];
            #pragma unroll
            for (int k = 0; k < j; ++k) v -= L[i][k] * L[j][k];
            L[i][j] = v * dinv[j];
        }
    }

    float x[DIM];
    #pragma unroll
    for (int i = 0; i < DIM; ++i) x[i] = 1.0f;

    #pragma unroll 1
    for (int it = 0; it < POW_ITERS; ++it) {
        float y[DIM], z[DIM];
        #pragma unroll
        for (int i = 0; i < DIM; ++i) {          // forward solve L y = x
            float v = x[i];
            #pragma unroll
            for (int k = 0; k < i; ++k) v -= L[i][k] * y[k];
            y[i] = v * dinv[i];
        }
        #pragma unroll
        for (int i = DIM - 1; i >= 0; --i) {     // backward solve L^T z = y
            float v = y[i];
            #pragma unroll
            for (int k = i + 1; k < DIM; ++k) v -= L[k][i] * z[k];
            z[i] = v * dinv[i];
        }
        float nrm = 0.0f;
        #pragma unroll
        for (int i = 0; i < DIM; ++i) nrm += z[i] * z[i];
        const float r = __builtin_amdgcn_rsqf(nrm);
        #pragma unroll
        for (int i = 0; i < DIM; ++i) x[i] = z[i] * r;
    }

    if (lane == 0) {
        #pragma unroll
        for (int i = 0; i < DIM; ++i) lds_f[i] = x[i];
    }
}

__global__ __launch_bounds__(32)
void fmat_irls_kernel(const float* __restrict__ A, float* __restrict__ out)
{
    __shared__ float lds_M[16 * 16];
    __shared__ float lds_f[DIM];

    const int    b    = blockIdx.x;
    const int    lane = threadIdx.x;
    const int    half = lane >> 4;
    const int    m    = lane & 15;
    const bool   mv   = (m < DIM);
    const float* Ab   = A + (size_t)b * (NPTS * DIM);

    // Initial: uniform weights -> plain normal equations
    v8f c = accumulate_gram<false>(Ab, lane, 0.0f);
    dump_accum(c, lds_M, lane);
    __syncthreads();
    solve_min_eigvec(lds_M, lds_f, lane);
    __syncthreads();

    #pragma unroll 1
    for (int it = 0; it < NUM_ITERS; ++it) {
        const float fm = mv ? lds_f[m] : 0.0f;   // per-lane f[col m]; 0 on padding lanes
        c = accumulate_gram<true>(Ab, lane, fm);
        dump_accum(c, lds_M, lane);
        __syncthreads();
        solve_min_eigvec(lds_M, lds_f, lane);
        __syncthreads();
    }

    if (half == 0 && mv)
        out[(size_t)b * DIM + m] = lds_f[m];
}

extern "C" void kernel_launch(void* const* d_in, const int* in_sizes, int n_in,
                              void* d_out, int out_size, void* d_ws, size_t ws_size,
                              hipStream_t stream)
{
    const float* A   = (const float*)d_in[0];
    float*       out = (float*)d_out;
    const int    B   = in_sizes[0] / (NPTS * DIM);   // 2048

    fmat_irls_kernel<<<dim3(B), dim3(32), 0, stream>>>(A, out);
}